// SparseGCNLayer_6373731467788
// MI455X (gfx1250) — compile-verified
//
#include <hip/hip_runtime.h>
#include <hip/hip_bf16.h>

typedef __attribute__((ext_vector_type(2))) float v2f;
typedef __attribute__((ext_vector_type(8))) float v8f;

#define GCN_D 128

// ---------------------------------------------------------------------------
// K0: out = 0, deg = 1.0  (deg starts at 1.0 == self-loop weight pre-added)
// ---------------------------------------------------------------------------
__global__ void gcn_init_kernel(float* __restrict__ out, float* __restrict__ deg,
                                long long n_out, int n_deg) {
    long long i = (long long)blockIdx.x * blockDim.x + threadIdx.x;
    long long stride = (long long)gridDim.x * blockDim.x;
    for (long long j = i; j < n_out; j += stride) out[j] = 0.0f;
    for (long long j = i; j < n_deg; j += stride) deg[j] = 1.0f;
}

// ---------------------------------------------------------------------------
// K1: deg[row[e]] += w[e]   (rows are edge_index[0, :] = first E entries)
// ---------------------------------------------------------------------------
__global__ void gcn_degree_kernel(const int* __restrict__ ei,
                                  const float* __restrict__ ew,
                                  float* __restrict__ deg, int E) {
    int i = blockIdx.x * blockDim.x + threadIdx.x;
    int stride = gridDim.x * blockDim.x;
    for (int e = i; e < E; e += stride)
        atomicAdd(deg + ei[e], ew[e]);
}

// ---------------------------------------------------------------------------
// K2: dis[i] = rsqrt(max(deg[i], 1e-12))   (in place)
// ---------------------------------------------------------------------------
__global__ void gcn_rsqrt_kernel(float* __restrict__ deg, int N) {
    int i = blockIdx.x * blockDim.x + threadIdx.x;
    if (i < N) deg[i] = rsqrtf(fmaxf(deg[i], 1e-12f));
}

// ---------------------------------------------------------------------------
// K3: h = x @ W^T + b  via V_WMMA_F32_16X16X4_F32 (fp32-exact matrix path)
//   grid.x = N/16 row tiles; block = 256 threads = 8 waves; wave w -> col tile w.
//   A (16x4 f32) per ISA layout: lanes 0-15 hold row M=lane, K = k0,k0+1;
//                                lanes 16-31 hold row M=lane-16, K = k0+2,k0+3.
//   B (4x16)  mirrored: lane supplies column n = lane&15, same K split.
//   C/D (16x16 f32): VGPR v, lanes 0-15 -> M=v, lanes 16-31 -> M=v+8, N=lane&15.
// ---------------------------------------------------------------------------
__global__ __launch_bounds__(256) void gcn_gemm_wmma_kernel(
    const float* __restrict__ x, const float* __restrict__ W,
    const float* __restrict__ b, float* __restrict__ h, int N) {
    const int wave    = threadIdx.x >> 5;          // 0..7  -> output col tile
    const int lane    = threadIdx.x & 31;
    const int half    = lane >> 4;                 // 0 or 1
    const int nl      = lane & 15;                 // col-in-tile / row-in-tile
    const int mbase   = blockIdx.x * 16;
    const int colbase = wave * 16;

    int mload = mbase + nl;                        // A row this lane feeds
    if (mload >= N) mload = N - 1;                 // clamp (stores are guarded)
    const int khalf = half * 2;                    // K sub-offset 0 or 2

    const float* __restrict__ xrow = x + (size_t)mload * GCN_D + khalf;
    const float* __restrict__ wrow = W + (size_t)(colbase + nl) * GCN_D + khalf;

    const float bn = b[colbase + nl];              // bias depends only on column
    v8f c = { bn, bn, bn, bn, bn, bn, bn, bn };

#pragma unroll
    for (int k0 = 0; k0 < GCN_D; k0 += 4) {
        v2f av = *(const v2f*)(xrow + k0);
        v2f bv = *(const v2f*)(wrow + k0);
        c = __builtin_amdgcn_wmma_f32_16x16x4_f32(
            /*neg_a=*/false, av, /*neg_b=*/false, bv,
            /*c_mod=*/(short)0, c, /*reuse_a=*/false, /*reuse_b=*/false);
    }

#pragma unroll
    for (int v = 0; v < 8; ++v) {
        const int mm = mbase + v + half * 8;
        if (mm < N)
            h[(size_t)mm * GCN_D + colbase + nl] = c[v];
    }
}

// ---------------------------------------------------------------------------
// K4: scatter  out[row] += h[col] * (dis[row] * w * dis[col])
//   Messages 0..E-1 are edges; messages E..E+N-1 are self loops (w = 1).
//   32 lanes per message: float4 gather of h (L2-resident), 4 f32 atomics.
// ---------------------------------------------------------------------------
__global__ __launch_bounds__(256) void gcn_scatter_kernel(
    const int* __restrict__ ei, const float* __restrict__ ew,
    const float* __restrict__ dis, const float* __restrict__ h,
    float* __restrict__ out, int E, int N) {
    long long gid = (long long)blockIdx.x * blockDim.x + threadIdx.x;
    long long msg = gid >> 5;
    const int lane = (int)(gid & 31);
    const long long total = (long long)E + N;
    if (msg >= total) return;

    int r, cidx; float wv;
    if (msg < E) {
        r    = ei[msg];
        cidx = ei[(long long)E + msg];
        wv   = ew[msg];
    } else {
        r = cidx = (int)(msg - E);
        wv = 1.0f;
    }
    const float norm = dis[r] * wv * dis[cidx];

    const float4 h4 = *(const float4*)(h + (size_t)cidx * GCN_D + lane * 4);
    float* op = out + (size_t)r * GCN_D + lane * 4;
    atomicAdd(op + 0, h4.x * norm);
    atomicAdd(op + 1, h4.y * norm);
    atomicAdd(op + 2, h4.z * norm);
    atomicAdd(op + 3, h4.w * norm);
}

// ---------------------------------------------------------------------------
// launch
// ---------------------------------------------------------------------------
extern "C" void kernel_launch(void* const* d_in, const int* in_sizes, int n_in,
                              void* d_out, int out_size, void* d_ws, size_t ws_size,
                              hipStream_t stream) {
    const float* x  = (const float*)d_in[0];   // [N,128]
    const float* W  = (const float*)d_in[1];   // [128,128]
    const float* b  = (const float*)d_in[2];   // [128]
    const int*   ei = (const int*)d_in[3];     // [2,E] row-major: rows then cols
    const float* ew = (const float*)d_in[4];   // [E]

    const int N = in_sizes[0] / GCN_D;
    const int E = in_sizes[4];

    float* out = (float*)d_out;
    float* h   = (float*)d_ws;                         // N*128 floats
    float* deg = h + (size_t)N * GCN_D;                // N floats (deg -> dis)

    // K0: zero output, deg = 1.0 (self loop)
    {
        long long n_out = (long long)N * GCN_D;
        int blocks = 2048;
        gcn_init_kernel<<<blocks, 256, 0, stream>>>(out, deg, n_out, N);
    }
    // K1: degree accumulation over edges
    gcn_degree_kernel<<<(E + 255) / 256, 256, 0, stream>>>(ei, ew, deg, E);
    // K2: deg -> deg^{-1/2}
    gcn_rsqrt_kernel<<<(N + 255) / 256, 256, 0, stream>>>(deg, N);
    // K3: h = x W^T + b   (WMMA f32)
    gcn_gemm_wmma_kernel<<<(N + 15) / 16, 256, 0, stream>>>(x, W, b, h, N);
    // K4: normalized gather-scatter aggregation
    {
        long long total   = (long long)E + N;           // messages
        long long threads = total * 32;
        unsigned blocks   = (unsigned)((threads + 255) / 256);
        gcn_scatter_kernel<<<blocks, 256, 0, stream>>>(ei, ew, deg, h, out, E, N);
    }
}